// BuddyPool_52664888983643
// MI455X (gfx1250) — compile-verified
//
#include <hip/hip_runtime.h>

// ---------------------------------------------------------------------------
// BuddyPool for MI455X (gfx1250, wave32).
// HBM-bound: patches (256 MB) streamed once => ~11 us floor at 23.3 TB/s.
// Phase 1 (512 blocks): sim via V_WMMA_F32_16X16X4_F32 with patches as the
//   A matrix (clean 8B/lane global_load_b64, zero address duplication) and
//   cue as the B matrix (one aligned ds_load_b64 from a [d0,d2,d1,d3]-
//   permuted LDS image). Garbage B columns (k>=5) only pollute accumulator
//   lanes we never read, so no masking/branching in the WMMA loop.
//   Per-wave first-max argmax partials -> d_ws.
// Phase 2 (256 blocks): reduce 64 partials per (b,k), fused clamped 3x3
//   ROI average (L2-resident re-reads).
// ---------------------------------------------------------------------------

typedef float v2f __attribute__((ext_vector_type(2)));
typedef float v8f __attribute__((ext_vector_type(8)));

constexpr int Bn = 64;
constexpr int Kn = 5;
constexpr int Dn = 1024;
constexpr int Hn = 32;
constexpr int Wn = 32;
constexpr int Pn = Hn * Wn;                    // 1024 positions
constexpr int WAVES = 8;                       // 256 threads = 8 wave32
constexpr int THREADS = WAVES * 32;
constexpr int TILES = Pn / 16;                 // 64 tiles of 16 positions
constexpr int GROUPS = TILES / WAVES;          // 8 blocks per batch (phase 1)
constexpr int NPART = Bn * Kn * TILES;         // partial-result count

__global__ __launch_bounds__(THREADS)
void sim_argmax_kernel(const float* __restrict__ cue,
                       const float* __restrict__ patches,
                       float* __restrict__ ws_val,
                       int* __restrict__ ws_idx) {
  __shared__ float s_cue[Kn * Dn];   // 20 KB, permuted [d0,d2,d1,d3] per quad

  const int g   = blockIdx.x;        // tile group 0..7
  const int b   = blockIdx.y;        // batch
  const int tid = threadIdx.x;
  const float* cue_b = cue + (size_t)b * Kn * Dn;
  const float* pat_b = patches + (size_t)b * Pn * Dn;

  // Stage cue into LDS with within-quad permutation so that for lane-half h,
  // the two B elements {d+h, d+2+h} sit adjacent at offset d + 2*h.
  for (int i = tid; i < Kn * Dn; i += THREADS) {
    const int r = i >> 10;
    const int j = i & (Dn - 1);
    const int pj = (j & ~3) | ((j & 1) << 1) | ((j & 2) >> 1);
    s_cue[r * Dn + pj] = cue_b[i];
  }
  __syncthreads();

  const int wave = tid >> 5;
  const int lane = tid & 31;
  const int half = lane >> 4;                  // 0: lanes 0-15, 1: 16-31
  const int l16  = lane & 15;
  const int t    = g * WAVES + wave;           // this wave's tile
  const int p0   = t * 16;

  // A: patches rows. Lane holds row M=l16, K elements {2*half, 2*half+1}.
  const float* aptr = pat_b + (size_t)(p0 + l16) * Dn + 2 * half;
  // B: cue columns N=l16 (k index). Clamp row: columns N>=Kn are never read.
  const int row = (l16 < Kn) ? l16 : 0;
  const float* bptr = &s_cue[row * Dn + 2 * half];

  v8f acc = {0.f, 0.f, 0.f, 0.f, 0.f, 0.f, 0.f, 0.f};

#pragma unroll 8
  for (int d = 0; d < Dn; d += 4) {
    const v2f am = *(const v2f*)(aptr + d);    // global_load_b64, no dup
    const v2f bm = *(const v2f*)(bptr + d);    // ds_load_b64 (permuted image)
    acc = __builtin_amdgcn_wmma_f32_16x16x4_f32(
        /*neg_a=*/false, am, /*neg_b=*/false, bm,
        /*c_mod=*/(short)0, acc, /*reuse_a=*/false, /*reuse_b=*/false);
  }

  // acc[j] on lane L = sim[p0 + j + 8*half][k = l16]. First-max over the
  // 16 positions of this tile, computed for all k in parallel across lanes.
  float bv = -3.402823466e38f;
  int   bi = 0x7fffffff;
#pragma unroll
  for (int j = 0; j < 8; ++j) {
    const float v = acc[j];
    const int pos = p0 + j + 8 * half;
    if (v > bv || (v == bv && pos < bi)) { bv = v; bi = pos; }
  }
  {
    const float v2 = __shfl_xor(bv, 16, 32);
    const int   i2 = __shfl_xor(bi, 16, 32);
    if (v2 > bv || (v2 == bv && i2 < bi)) { bv = v2; bi = i2; }
  }
  if (lane < Kn) {                             // lanes 0..4 hold k=0..4
    const int o = (b * Kn + lane) * TILES + t;
    ws_val[o] = bv;
    ws_idx[o] = bi;
  }
}

__global__ __launch_bounds__(THREADS)
void reduce_roi_kernel(const float* __restrict__ patches,
                       const float* __restrict__ ws_val,
                       const int* __restrict__ ws_idx,
                       float* __restrict__ out) {
  __shared__ int s_idx[Kn];

  const int chunk = blockIdx.x;                // 0..3 quarter of (k,d) space
  const int b     = blockIdx.y;
  const int tid   = threadIdx.x;
  const float* pat_b = patches + (size_t)b * Pn * Dn;

  if (tid < Kn) {
    float bv = -3.402823466e38f;
    int   bi = 0x7fffffff;
    const int base = (b * Kn + tid) * TILES;
    for (int u = 0; u < TILES; ++u) {
      const float v = ws_val[base + u];
      const int   i = ws_idx[base + u];
      if (v > bv || (v == bv && i < bi)) { bv = v; bi = i; }
    }
    s_idx[tid] = bi;
  }
  __syncthreads();

  const int span = (Kn * Dn) / 4;              // 1280 elements per chunk
  const int e0 = chunk * span;
  for (int e = e0 + tid; e < e0 + span; e += THREADS) {
    const int k = e >> 10;                     // e / Dn
    const int d = e & (Dn - 1);
    const int idx = s_idx[k];
    const int h = idx >> 5;
    const int w = idx & (Wn - 1);
    const int r0 = (h - 1 > 0) ? h - 1 : 0;
    const int r1 = (h + 1 < Hn - 1) ? h + 1 : Hn - 1;
    const int c0 = (w - 1 > 0) ? w - 1 : 0;
    const int c1 = (w + 1 < Wn - 1) ? w + 1 : Wn - 1;
    float s = 0.f;
    for (int r = r0; r <= r1; ++r)
      for (int c = c0; c <= c1; ++c)
        s += pat_b[(size_t)((r << 5) | c) * Dn + d];
    const float cnt = (float)((r1 - r0 + 1) * (c1 - c0 + 1));
    out[(size_t)b * Kn * Dn + e] = s / cnt;
  }
}

extern "C" void kernel_launch(void* const* d_in, const int* in_sizes, int n_in,
                              void* d_out, int out_size, void* d_ws, size_t ws_size,
                              hipStream_t stream) {
  (void)in_sizes; (void)n_in; (void)out_size; (void)ws_size;
  const float* cue     = (const float*)d_in[0];   // (64, 5, 1024) f32
  const float* patches = (const float*)d_in[1];   // (64, 32, 32, 1024) f32
  float* out = (float*)d_out;                     // (64, 5, 1024) f32

  float* ws_val = (float*)d_ws;                            // 20480 floats
  int*   ws_idx = (int*)((char*)d_ws + NPART * sizeof(float)); // 20480 ints

  sim_argmax_kernel<<<dim3(GROUPS, Bn), THREADS, 0, stream>>>(
      cue, patches, ws_val, ws_idx);
  reduce_roi_kernel<<<dim3(4, Bn), THREADS, 0, stream>>>(
      patches, ws_val, ws_idx, out);
}